// End2End_30073361006971
// MI455X (gfx1250) — compile-verified
//
#include <hip/hip_runtime.h>
#include <stdint.h>

// ---------------------------------------------------------------------------
// YOLO NMS post-processing for MI455X (gfx1250, wave32).
//   K0: zero hist/counters/keys
//   K1: TDM-staged score + 65536-bin histogram of conf float bits
//   K2: find threshold bin s.t. count(score >= bin) >= 30000  (1 WG)
//   K3: TDM-staged compaction (key = confBits<<32 | ~idx) via atomic counter
//   K4: single-WG: TDM DMA keys (256KB) -> LDS, bitonic sort descending,
//       gather top-30000, sequential greedy NMS, word-owned alive bitmask.
// ---------------------------------------------------------------------------

#define N_PRED     1048576
#define ROW        7
#define CONF_TH    0.25f
#define IOU_TH     0.45f
#define MAX_WH     4096.0f
#define MAX_NMS    30000
#define MAX_DET    300
#define CAP        32768           // sort capacity (pow2), 256 KB of 8B keys
#define NBINS      65536
#define BIN_BASE   0x3E8000u       // (bits of 0.25f) >> 8
#define SORT_T     1024
#define TILE_T     256             // threads (and rows) per streaming tile
#define AWORDS     ((MAX_NMS + 31) / 32)   // 938

typedef unsigned int uint32x4 __attribute__((ext_vector_type(4)));
typedef int          int32x4v __attribute__((ext_vector_type(4)));
typedef int          int32x8v __attribute__((ext_vector_type(8)));

#if __has_builtin(__builtin_amdgcn_tensor_load_to_lds)
#define HAVE_TDM 1
#endif

#ifdef HAVE_TDM
// 1-D Tensor Data Mover DMA: nelem elements (dsz_code: 2 => 4B, 3 => 8B) from
// global 'ga' into LDS offset 'lds_off'. Caller must be a single wave (TDM
// ignores EXEC; gate with a wave-uniform branch). Waits TENSORcnt==0.
__device__ __forceinline__ void tdm_load_1d(unsigned lds_off, unsigned long long ga,
                                            unsigned nelem, unsigned dsz_code) {
  uint32x4 g0;
  g0[0] = 1u;                                                  // count=1 (valid D#)
  g0[1] = lds_off;                                             // lds_addr
  g0[2] = (unsigned)ga;                                        // global_addr[31:0]
  g0[3] = (unsigned)((ga >> 32) & 0x01FFFFFFu) | (2u << 30);   // addr[56:32] | type=2
  int32x8v g1;
  g1[0] = (int)(dsz_code << 16);              // wg_mask=0, data_size, no flags
  g1[1] = (int)((nelem & 0xFFFFu) << 16);     // tensor_dim0[15:0] at bits 63:48
  g1[2] = (int)((nelem >> 16) & 0xFFFFu) | (1 << 16); // tdim0[31:16], tensor_dim1=1
  g1[3] = (int)((nelem & 0xFFFFu) << 16);     // tile_dim0 at bits 127:112 (<=65535)
  g1[4] = 0;                                  // tile_dim1=0, tile_dim2=0 (unused)
  g1[5] = (int)nelem;                         // tensor_dim0_stride[31:0]
  g1[6] = 0;
  g1[7] = 0;
  int32x4v gz = {0, 0, 0, 0};
#if defined(__clang_major__) && (__clang_major__ >= 23)
  int32x8v gz8 = {0, 0, 0, 0, 0, 0, 0, 0};
  __builtin_amdgcn_tensor_load_to_lds(g0, g1, gz, gz, gz8, 0);
#else
  __builtin_amdgcn_tensor_load_to_lds(g0, g1, gz, gz, 0);
#endif
#if __has_builtin(__builtin_amdgcn_s_wait_tensorcnt)
  __builtin_amdgcn_s_wait_tensorcnt(0);
#endif
}
#endif

__device__ __forceinline__ void score_row(const float* p,
                                          float& conf, int& bin, bool& valid) {
  float obj = p[4];
  float s0 = p[5] * obj;          // reference: cls = pred[:,5:7] * obj, then max
  float s1 = p[6] * obj;
  conf = fmaxf(s0, s1);
  valid = (obj > CONF_TH) && (conf > CONF_TH);
  // conf in (0.25, 1) when valid -> positive-float bits are order-preserving
  bin = (int)((__float_as_uint(conf) >> 8) - BIN_BASE);   // [0, 65535] when valid
}

__global__ void k_zero(unsigned* __restrict__ p, int n) {
  int i = blockIdx.x * blockDim.x + threadIdx.x;
  if (i < n) p[i] = 0u;
}

// Stage one 256-row tile (7168 B) of predictions into LDS via TDM, return
// pointer to this thread's row.
__device__ __forceinline__ const float* stage_tile(const float* __restrict__ pred,
                                                   float* tile /* LDS */) {
  const int tid = threadIdx.x;
  const size_t base = (size_t)blockIdx.x * (TILE_T * ROW);
#ifdef HAVE_TDM
  if (tid < 32) {    // wave 0 issues the DMA; uniform branch (TDM ignores EXEC)
    tdm_load_1d((unsigned)(uintptr_t)tile,
                (unsigned long long)(uintptr_t)(pred + base),
                TILE_T * ROW, 2u /* 4-byte elements */);
  }
#else
  for (int i = tid; i < TILE_T * ROW; i += TILE_T) tile[i] = pred[base + i];
#endif
  __syncthreads();
  return &tile[tid * ROW];
}

__global__ __launch_bounds__(TILE_T) void k_hist(const float* __restrict__ pred,
                                                 unsigned* __restrict__ hist) {
  __shared__ float tile[TILE_T * ROW];
  const float* row = stage_tile(pred, tile);
  float conf; int bin; bool valid;
  score_row(row, conf, bin, valid);
  if (valid) atomicAdd(&hist[min(bin, NBINS - 1)], 1u);
}

// counters[0] = candidate count, counters[1] = threshold bin
__global__ __launch_bounds__(1024) void k_thresh(const unsigned* __restrict__ hist,
                                                 unsigned* __restrict__ counters) {
  __shared__ unsigned csum[1024];
  __shared__ unsigned s_cidx, s_need;
  const int t = threadIdx.x;
  const int CHUNK = NBINS / 1024;   // 64 bins per thread
  unsigned s = 0;
  for (int i = 0; i < CHUNK; ++i) s += hist[t * CHUNK + i];
  csum[t] = s;
  __syncthreads();
  if (t == 0) {
    unsigned acc = 0, need = 0; int cidx = -1;
    for (int c = 1023; c >= 0; --c) {          // walk from the top score down
      if (acc + csum[c] >= (unsigned)MAX_NMS) { cidx = c; need = MAX_NMS - acc; break; }
      acc += csum[c];
    }
    s_cidx = (unsigned)cidx; s_need = need;
    if (cidx < 0) counters[1] = 0;             // fewer than MAX_NMS valid: keep all
  }
  __syncthreads();
  if ((int)s_cidx >= 0 && t == (int)s_cidx) {
    unsigned acc = 0; int base = t * CHUNK, thr = base;
    for (int i = CHUNK - 1; i >= 0; --i) {
      acc += hist[base + i];
      if (acc >= s_need) { thr = base + i; break; }
    }
    counters[1] = (unsigned)thr;
  }
}

__global__ __launch_bounds__(TILE_T) void k_compact(const float* __restrict__ pred,
                                                    unsigned* __restrict__ counters,
                                                    unsigned long long* __restrict__ keys) {
  __shared__ float tile[TILE_T * ROW];
  const float* row = stage_tile(pred, tile);
  const int i = blockIdx.x * TILE_T + threadIdx.x;
  float conf; int bin; bool valid;
  score_row(row, conf, bin, valid);
  unsigned thr = counters[1];
  if (valid && (unsigned)bin >= thr) {
    unsigned slot = atomicAdd(&counters[0], 1u);
    if (slot < CAP) {
      // descending sort key: score bits high, ~idx low (ties -> lower idx first)
      keys[slot] = ((unsigned long long)__float_as_uint(conf) << 32)
                 | (unsigned long long)(~(unsigned)i);
    }
  }
}

__global__ __launch_bounds__(SORT_T) void k_sort_nms(
    const float* __restrict__ pred,
    const unsigned long long* __restrict__ keys,
    float4* __restrict__ boxOff, float4* __restrict__ boxOrig,
    float* __restrict__ areaA, float* __restrict__ confA, float* __restrict__ jA,
    float* __restrict__ out) {
  // 256 KB key array + bitmask: only possible because CDNA5 gives 320 KB LDS/WGP.
  __shared__ unsigned long long skeys[CAP];
  __shared__ unsigned alive[AWORDS];
  __shared__ int    s_minWord;
  __shared__ float4 s_box;
  __shared__ float  s_area;

  const int tid = threadIdx.x;

  // ---- Stage A: bulk global->LDS fill of the key buffer via the TDM
  // (one 1-D descriptor, 32768 x 8B), issued by wave 0 only.
#ifdef HAVE_TDM
  if (tid < 32) {
    tdm_load_1d((unsigned)(uintptr_t)&skeys[0],
                (unsigned long long)(uintptr_t)keys,
                CAP, 3u /* 8-byte elements */);
  }
#else
  for (int t = tid; t < CAP; t += SORT_T) skeys[t] = keys[t];
#endif
  __syncthreads();

  // ---- Stage B: bitonic sort, descending (zero-padded keys sink to the end).
  for (unsigned k = 2; k <= CAP; k <<= 1) {
    for (unsigned j = k >> 1; j > 0; j >>= 1) {
      for (unsigned i = tid; i < CAP; i += SORT_T) {
        unsigned p = i ^ j;
        if (p > i) {
          unsigned long long a = skeys[i], b = skeys[p];
          bool up = (i & k) == 0;
          if (up ? (a < b) : (a > b)) { skeys[i] = b; skeys[p] = a; }
        }
      }
      __syncthreads();
    }
  }

  // ---- Stage C: gather top-30000, build class-offset boxes + alive bitmask.
  for (int t = tid; t < AWORDS; t += SORT_T) alive[t] = 0u;
  __syncthreads();
  for (int t = tid; t < MAX_NMS; t += SORT_T) {
    unsigned long long key = skeys[t];
    if ((unsigned)(key >> 32) != 0u) {               // real candidate (pad key == 0)
      unsigned idx = ~(unsigned)(key & 0xFFFFFFFFull);
      const float* p = pred + (size_t)idx * ROW;
      float x = p[0], y = p[1], w = p[2], h = p[3], obj = p[4];
      float s0 = p[5] * obj, s1 = p[6] * obj;
      float conf = fmaxf(s0, s1);
      float jj = (s1 > s0) ? 1.0f : 0.0f;
      float x0 = x - w * 0.5f, y0 = y - h * 0.5f;
      float x1 = x + w * 0.5f, y1 = y + h * 0.5f;
      float off = jj * MAX_WH;                       // class-offset for IoU only
      float4 bo = make_float4(x0 + off, y0 + off, x1 + off, y1 + off);
      boxOff[t]  = bo;
      boxOrig[t] = make_float4(x0, y0, x1, y1);
      areaA[t]   = (bo.z - bo.x) * (bo.w - bo.y);    // area from offset boxes (ref)
      confA[t]   = conf;
      jA[t]      = jj;
      atomicOr(&alive[t >> 5], 1u << (t & 31));
    }
  }
  __syncthreads();

  // ---- Stage D: sequential greedy NMS. Sorted desc => argmax == first alive.
  // Each alive-word is owned by exactly one thread: no atomics in the scan.
  for (int step = 0; step < MAX_DET; ++step) {
    if (tid == 0) s_minWord = 0x7FFFFFFF;
    __syncthreads();                                  // orders prior word stores
    if (tid < AWORDS && alive[tid] != 0u) atomicMin(&s_minWord, tid);
    __syncthreads();
    int mw = s_minWord;
    if (mw == 0x7FFFFFFF) {                           // exhausted: zero rows, ok=0
      for (int r = step + tid; r < MAX_DET; r += SORT_T) {
        float* row = out + r * 6;
        row[0] = row[1] = row[2] = row[3] = row[4] = row[5] = 0.0f;
        out[MAX_DET * 6 + r] = 0.0f;
      }
      break;                                          // uniform across the WG
    }
    if (tid == 0) {
      int sel = (mw << 5) + (__ffs(alive[mw]) - 1);
      alive[mw] &= ~(1u << (sel & 31));               // self-suppress (IoU==1)
      s_box  = boxOff[sel];
      s_area = areaA[sel];
      float4 o = boxOrig[sel];
      float* row = out + step * 6;                    // dets row (un-offset box)
      row[0] = o.x; row[1] = o.y; row[2] = o.z; row[3] = o.w;
      row[4] = confA[sel]; row[5] = jA[sel];
      out[MAX_DET * 6 + step] = 1.0f;                 // ok flag
    }
    __syncthreads();
    if (tid < AWORDS) {
      unsigned wbits = alive[tid];
      if (wbits != 0u) {                              // dead 32-entry chunks skipped
        const float4 b = s_box; const float ba = s_area;
        const int base = tid << 5;
        __builtin_prefetch(&boxOff[base], 0, 0);      // global_prefetch_b8
        unsigned rem = wbits, clear = 0u;
        while (rem) {
          int bit = __ffs(rem) - 1; rem &= rem - 1u;
          int t = base + bit;
          float4 c = boxOff[t];
          float ltx = fmaxf(b.x, c.x), lty = fmaxf(b.y, c.y);
          float rbx = fminf(b.z, c.z), rby = fminf(b.w, c.w);
          float iw = fmaxf(rbx - ltx, 0.0f), ih = fmaxf(rby - lty, 0.0f);
          float inter = iw * ih;
          float iou = inter / (ba + areaA[t] - inter);
          if (iou > IOU_TH) clear |= 1u << bit;
        }
        if (clear) alive[tid] = wbits & ~clear;       // sole owner: plain store
      }
    }
    // next iteration's first __syncthreads orders these stores before reads
  }
}

extern "C" void kernel_launch(void* const* d_in, const int* in_sizes, int n_in,
                              void* d_out, int out_size, void* d_ws, size_t ws_size,
                              hipStream_t stream) {
  (void)in_sizes; (void)n_in; (void)out_size; (void)ws_size;
  const float* pred = (const float*)d_in[0];
  float* out = (float*)d_out;
  char* ws = (char*)d_ws;

  // Workspace layout (~1.76 MB total)
  unsigned*            hist     = (unsigned*)(ws + 0);                // 262144 B
  unsigned*            counters = (unsigned*)(ws + 262144);           //    256 B
  unsigned long long*  keys     = (unsigned long long*)(ws + 262400); // 262144 B
  float4*              boxOff   = (float4*)(ws + 524544);             // 480000 B
  float4*              boxOrig  = (float4*)(ws + 1004544);            // 480000 B
  float*               areaA    = (float*)(ws + 1484544);             // 120000 B
  float*               confA    = (float*)(ws + 1604544);             // 120000 B
  float*               jA       = (float*)(ws + 1724544);             // 120000 B

  const int zeroN = (262144 + 256 + 262144) / 4;   // hist + counters + keys
  k_zero<<<(zeroN + 1023) / 1024, 1024, 0, stream>>>((unsigned*)ws, zeroN);
  k_hist<<<N_PRED / TILE_T, TILE_T, 0, stream>>>(pred, hist);
  k_thresh<<<1, 1024, 0, stream>>>(hist, counters);
  k_compact<<<N_PRED / TILE_T, TILE_T, 0, stream>>>(pred, counters, keys);
  k_sort_nms<<<1, SORT_T, 0, stream>>>(pred, keys, boxOff, boxOrig,
                                       areaA, confA, jA, out);
}